// GNNDir2encoder_12652973654571
// MI455X (gfx1250) — compile-verified
//
#include <hip/hip_runtime.h>
#include <math.h>

#define N_NODES  100000
#define N_EDGES  1600000
#define D_IN     128
#define NH       256
#define N_GRAPHS 256
#define EPSV     1e-5f

typedef __attribute__((ext_vector_type(2))) float v2f;
typedef __attribute__((ext_vector_type(8))) float v8f;

// ---------------------------------------------------------------------------
// CDNA5 fp32 WMMA: D(16x16) = A(16x4) * B(4x16) + C, wave32.
// A frag (lane l): rows m=l%16, ks = {koff, koff+1}, koff = (l>=16)?2:0
// B frag (lane l): cols n=l%16, same k pattern  (B[k][n] == W[n][k])
// C/D  (lane l, vgpr v): element [m = v + 8*(l/16)][n = l%16]
// ---------------------------------------------------------------------------
__device__ __forceinline__ v8f wmma4(v2f a, v2f b, v8f c) {
  return __builtin_amdgcn_wmma_f32_16x16x4_f32(false, a, false, b, (short)0, c,
                                               false, false);
}

// C[row0:+16, col0:+16] += A[row0:+16, 0:K] @ W[col0:+16, 0:K]^T
__device__ __forceinline__ void mm_loop(const float* A, int lda,
                                        const float* W, int ldw,
                                        int row0, int col0, int K,
                                        int lane, v8f& acc) {
  int m    = lane & 15;
  int koff = (lane >> 4) << 1;
  const float* ap = A + (size_t)(row0 + m) * lda + koff;
  const float* bp = W + (size_t)(col0 + m) * ldw + koff;   // n == m here
#pragma unroll 4
  for (int kk = 0; kk < K; kk += 4) {
    v2f a = *(const v2f*)(ap + kk);
    v2f b = *(const v2f*)(bp + kk);
    acc = wmma4(a, b, acc);
  }
}

// ------------------------- stage kernels -----------------------------------

// x = word_vec[idx_x]; agg = idx_w * x   (self-loop term)
__global__ void k_gather(const int* __restrict__ idx, const float* __restrict__ wv,
                         const float* __restrict__ idx_w,
                         float* __restrict__ X, float* __restrict__ AGG) {
  size_t t = (size_t)blockIdx.x * blockDim.x + threadIdx.x;
  if (t >= (size_t)N_NODES * 32) return;
  size_t i = t >> 5;
  int    c = (int)(t & 31);
  float  w = idx_w[0];
  float4 v = ((const float4*)(wv + (size_t)idx[i] * D_IN))[c];
  ((float4*)(X + i * D_IN))[c] = v;
  float4 s = {w * v.x, w * v.y, w * v.z, w * v.w};
  ((float4*)(AGG + i * D_IN))[c] = s;
}

// agg[dst] += w_e * x[src]   (one wave per edge, lane -> float4 chunk)
__global__ void k_edges(const int* __restrict__ src, const int* __restrict__ dst,
                        const float* __restrict__ ew,
                        const float* __restrict__ X, float* __restrict__ AGG) {
  size_t t = (size_t)blockIdx.x * blockDim.x + threadIdx.x;
  size_t e = t >> 5;
  if (e >= N_EDGES) return;
  int lane = (int)(t & 31);
  int s = src[e], d = dst[e];
  float w = ew[e];
  float4 v = ((const float4*)(X + (size_t)s * D_IN))[lane];
  float* ad = AGG + (size_t)d * D_IN + lane * 4;
  atomicAdd(ad + 0, w * v.x);
  atomicAdd(ad + 1, w * v.y);
  atomicAdd(ad + 2, w * v.z);
  atomicAdd(ad + 3, w * v.w);
}

// h = agg@Wrel^T + b_rel + x@Wroot^T   (in place over agg)
__global__ void k_gemm_h(const float* AGG, const float* __restrict__ X,
                         const float* __restrict__ w_rel,
                         const float* __restrict__ w_root,
                         const float* __restrict__ b_rel, float* H) {
  int lane = threadIdx.x & 31, wave = threadIdx.x >> 5;
  int row0 = blockIdx.x << 4;
  int col0 = wave << 4;                       // 8 waves * 16 = 128 cols
  v8f acc = {};
  mm_loop(AGG, 128, w_rel, 128, row0, col0, 128, lane, acc);
  mm_loop(X, 128, w_root, 128, row0, col0, 128, lane, acc);
  int m = lane & 15, mh = (lane >> 4) << 3;
  float bias = b_rel[col0 + m];
  __syncthreads();                            // finish all reads before in-place write
#pragma unroll
  for (int v = 0; v < 8; ++v)
    H[(size_t)(row0 + mh + v) * 128 + col0 + m] = acc[v] + bias;
}

// per-column sum / sumsq of h  (stats[0:128]=sum, [128:256]=sumsq)
__global__ void k_bn_stats(const float* __restrict__ H, float* __restrict__ st) {
  int c = threadIdx.x & 127;
  int rb = blockIdx.x * 2 + (threadIdx.x >> 7);
  int stride = gridDim.x * 2;
  float s = 0.f, ss = 0.f;
  for (int r = rb; r < N_NODES; r += stride) {
    float v = H[(size_t)r * 128 + c];
    s += v; ss += v * v;
  }
  atomicAdd(&st[c], s);
  atomicAdd(&st[128 + c], ss);
}

// enc1 = [tanh(bn(h)), x]
__global__ void k_enc1(const float* __restrict__ H, const float* __restrict__ X,
                       const float* __restrict__ st, const float* __restrict__ gam,
                       const float* __restrict__ bet, float* __restrict__ E1) {
  size_t t = (size_t)blockIdx.x * blockDim.x + threadIdx.x;
  size_t i = t >> 8;
  int    c = (int)(t & 255);
  float out;
  if (c < 128) {
    const float invN = 1.0f / (float)N_NODES;
    float mu  = st[c] * invN;
    float var = st[128 + c] * invN - mu * mu;
    out = tanhf(gam[c] * (H[i * 128 + c] - mu) * rsqrtf(var + EPSV) + bet[c]);
  } else {
    out = X[i * 128 + (c - 128)];
  }
  E1[t] = out;
}

// enc2 = sigmoid(enc1@fc1^T+b1) * tanh(enc1@fc2^T+b2)
__global__ void k_gemm_enc2(const float* __restrict__ E1,
                            const float* __restrict__ w1, const float* __restrict__ b1,
                            const float* __restrict__ w2, const float* __restrict__ b2,
                            float* __restrict__ E2) {
  int lane = threadIdx.x & 31, wave = threadIdx.x >> 5;
  int row0 = blockIdx.x << 4;
  int c0 = wave << 5, c1 = c0 + 16;           // 8 waves * 32 = 256 cols
  int m = lane & 15, koff = (lane >> 4) << 1;
  const float* ap  = E1 + (size_t)(row0 + m) * 256 + koff;
  const float* b1a = w1 + (size_t)(c0 + m) * 256 + koff;
  const float* b1b = w1 + (size_t)(c1 + m) * 256 + koff;
  const float* b2a = w2 + (size_t)(c0 + m) * 256 + koff;
  const float* b2b = w2 + (size_t)(c1 + m) * 256 + koff;
  v8f u0 = {}, u1 = {}, t0 = {}, t1 = {};
#pragma unroll 2
  for (int kk = 0; kk < 256; kk += 4) {
    v2f a = *(const v2f*)(ap + kk);
    u0 = wmma4(a, *(const v2f*)(b1a + kk), u0);
    u1 = wmma4(a, *(const v2f*)(b1b + kk), u1);
    t0 = wmma4(a, *(const v2f*)(b2a + kk), t0);
    t1 = wmma4(a, *(const v2f*)(b2b + kk), t1);
  }
  int mh = (lane >> 4) << 3;
  float bb10 = b1[c0 + m], bb11 = b1[c1 + m];
  float bb20 = b2[c0 + m], bb21 = b2[c1 + m];
#pragma unroll
  for (int v = 0; v < 8; ++v) {
    float s0 = 1.f / (1.f + expf(-(u0[v] + bb10)));
    float s1 = 1.f / (1.f + expf(-(u1[v] + bb11)));
    size_t r = (size_t)(row0 + mh + v) * 256;
    E2[r + c0 + m] = s0 * tanhf(t0[v] + bb20);
    E2[r + c1 + m] = s1 * tanhf(t1[v] + bb21);
  }
}

// pooled[g] += enc2[i]
__global__ void k_pool(const float* __restrict__ E2, const int* __restrict__ xb,
                       float* __restrict__ PL) {
  size_t t = (size_t)blockIdx.x * blockDim.x + threadIdx.x;
  if (t >= (size_t)N_NODES * 64) return;
  size_t i = t >> 6;
  int   c4 = (int)(t & 63);
  int    g = xb[i];
  float4 v = ((const float4*)(E2 + i * 256))[c4];
  float* p = PL + (size_t)g * 256 + c4 * 4;
  atomicAdd(p + 0, v.x);
  atomicAdd(p + 1, v.y);
  atomicAdd(p + 2, v.z);
  atomicAdd(p + 3, v.w);
}

// C[256,256] = A[256,256] @ W[256,256]^T + bias
__global__ void k_gemm_small(const float* __restrict__ A, const float* __restrict__ W,
                             const float* __restrict__ bias, float* __restrict__ C) {
  int lane = threadIdx.x & 31, wave = threadIdx.x >> 5;
  int row0 = blockIdx.x << 4;
  int c0 = wave << 5, c1 = c0 + 16;
  v8f a0 = {}, a1 = {};
  mm_loop(A, 256, W, 256, row0, c0, 256, lane, a0);
  mm_loop(A, 256, W, 256, row0, c1, 256, lane, a1);
  int m = lane & 15, mh = (lane >> 4) << 3;
  float bb0 = bias[c0 + m], bb1 = bias[c1 + m];
#pragma unroll
  for (int v = 0; v < 8; ++v) {
    size_t r = (size_t)(row0 + mh + v) * 256;
    C[r + c0 + m] = a0[v] + bb0;
    C[r + c1 + m] = a1[v] + bb1;
  }
}

// mean = (t - mu)/sqrt(var+eps) + meanbn_beta  (column stats over 256 rows)
__global__ void k_meanbn(const float* __restrict__ T, const float* __restrict__ mb,
                         float* __restrict__ out) {
  int j = threadIdx.x;
  float s = 0.f, ss = 0.f;
  for (int i = 0; i < 256; ++i) {
    float v = T[(size_t)i * 256 + j];
    s += v; ss += v * v;
  }
  float mu  = s * (1.f / 256.f);
  float var = ss * (1.f / 256.f) - mu * mu;
  float inv = rsqrtf(var + EPSV);
  for (int i = 0; i < 256; ++i)
    out[(size_t)i * 256 + j] = (T[(size_t)i * 256 + j] - mu) * inv + mb[j];
}

// logits = [enc1, pooled[g]] @ phi_w^T + phi_b   (K = 512 concat)
__global__ void k_gemm_phi(const float* __restrict__ E1, const float* __restrict__ PL,
                           const int* __restrict__ xb, const float* __restrict__ W,
                           const float* __restrict__ bias, float* __restrict__ L) {
  int lane = threadIdx.x & 31, wave = threadIdx.x >> 5;
  int row0 = blockIdx.x << 4;
  int c0 = wave << 5, c1 = c0 + 16;
  int m = lane & 15, koff = (lane >> 4) << 1;
  v8f a0 = {}, a1 = {};
  {
    const float* ap  = E1 + (size_t)(row0 + m) * 256 + koff;
    const float* bpa = W + (size_t)(c0 + m) * 512 + koff;
    const float* bpb = W + (size_t)(c1 + m) * 512 + koff;
#pragma unroll 2
    for (int kk = 0; kk < 256; kk += 4) {
      v2f a = *(const v2f*)(ap + kk);
      a0 = wmma4(a, *(const v2f*)(bpa + kk), a0);
      a1 = wmma4(a, *(const v2f*)(bpb + kk), a1);
    }
  }
  {
    int g = xb[row0 + m];                      // per-lane graph id
    const float* ap  = PL + (size_t)g * 256 + koff;
    const float* bpa = W + (size_t)(c0 + m) * 512 + 256 + koff;
    const float* bpb = W + (size_t)(c1 + m) * 512 + 256 + koff;
#pragma unroll 2
    for (int kk = 0; kk < 256; kk += 4) {
      v2f a = *(const v2f*)(ap + kk);
      a0 = wmma4(a, *(const v2f*)(bpa + kk), a0);
      a1 = wmma4(a, *(const v2f*)(bpb + kk), a1);
    }
  }
  int mh = (lane >> 4) << 3;
  float bb0 = bias[c0 + m], bb1 = bias[c1 + m];
#pragma unroll
  for (int v = 0; v < 8; ++v) {
    size_t r = (size_t)(row0 + mh + v) * 256;
    L[r + c0 + m] = a0[v] + bb0;
    L[r + c1 + m] = a1[v] + bb1;
  }
}

// row softmax over 256 cols, one wave per row, in place
__global__ void k_softmax(float* __restrict__ phi) {
  int row = blockIdx.x * 8 + (threadIdx.x >> 5);
  if (row >= N_NODES) return;
  int lane = threadIdx.x & 31;
  float4* p = (float4*)(phi + (size_t)row * 256);
  float4 v0 = p[lane], v1 = p[lane + 32];
  float mx = fmaxf(fmaxf(fmaxf(v0.x, v0.y), fmaxf(v0.z, v0.w)),
                   fmaxf(fmaxf(v1.x, v1.y), fmaxf(v1.z, v1.w)));
#pragma unroll
  for (int o = 16; o > 0; o >>= 1) mx = fmaxf(mx, __shfl_xor(mx, o, 32));
  v0.x = expf(v0.x - mx); v0.y = expf(v0.y - mx);
  v0.z = expf(v0.z - mx); v0.w = expf(v0.w - mx);
  v1.x = expf(v1.x - mx); v1.y = expf(v1.y - mx);
  v1.z = expf(v1.z - mx); v1.w = expf(v1.w - mx);
  float s = v0.x + v0.y + v0.z + v0.w + v1.x + v1.y + v1.z + v1.w;
#pragma unroll
  for (int o = 16; o > 0; o >>= 1) s += __shfl_xor(s, o, 32);
  float inv = 1.f / s;
  v0.x *= inv; v0.y *= inv; v0.z *= inv; v0.w *= inv;
  v1.x *= inv; v1.y *= inv; v1.z *= inv; v1.w *= inv;
  p[lane] = v0; p[lane + 32] = v1;
}

// --------------------------- launcher --------------------------------------
extern "C" void kernel_launch(void* const* d_in, const int* in_sizes, int n_in,
                              void* d_out, int out_size, void* d_ws, size_t ws_size,
                              hipStream_t stream) {
  const int*   idx_x    = (const int*)d_in[0];
  const float* idx_w    = (const float*)d_in[1];
  const int*   x_batch  = (const int*)d_in[2];
  const int*   eidx     = (const int*)d_in[3];
  const float* eweight  = (const float*)d_in[4];
  const float* word_vec = (const float*)d_in[5];
  const float* w_rel    = (const float*)d_in[6];
  const float* b_rel    = (const float*)d_in[7];
  const float* w_root   = (const float*)d_in[8];
  const float* bn1_g    = (const float*)d_in[9];
  const float* bn1_b    = (const float*)d_in[10];
  const float* fc1_w    = (const float*)d_in[11];
  const float* fc1_b    = (const float*)d_in[12];
  const float* fc2_w    = (const float*)d_in[13];
  const float* fc2_b    = (const float*)d_in[14];
  const float* mean_w   = (const float*)d_in[15];
  const float* mean_b   = (const float*)d_in[16];
  const float* mbeta    = (const float*)d_in[17];
  const float* lv_w     = (const float*)d_in[18];
  const float* lv_b     = (const float*)d_in[19];
  const float* phi_w    = (const float*)d_in[20];
  const float* phi_b    = (const float*)d_in[21];

  float* X   = (float*)d_ws;                         // [N,128]
  float* AGG = X + (size_t)N_NODES * D_IN;           // [N,128]  (h in place)
  float* E1  = AGG + (size_t)N_NODES * D_IN;         // [N,256]
  float* E2  = X;                                    // [N,256] overlays X+AGG (dead)
  float* PL  = E1 + (size_t)N_NODES * NH;            // [256,256]
  float* ST  = PL + 65536;                           // bn stats (256 used)
  float* TM  = ST + 1024;                            // [256,256] pre-bn mean

  float* out_mean = (float*)d_out;
  float* out_lv   = out_mean + 65536;
  float* out_phi  = out_lv + 65536;

  hipMemsetAsync(PL, 0, (65536 + 1024) * sizeof(float), stream);

  k_gather<<<(N_NODES * 32) / 256, 256, 0, stream>>>(idx_x, word_vec, idx_w, X, AGG);
  k_edges<<<(N_EDGES * 32) / 256, 256, 0, stream>>>(eidx, eidx + N_EDGES, eweight, X, AGG);
  k_gemm_h<<<N_NODES / 16, 256, 0, stream>>>(AGG, X, w_rel, w_root, b_rel, AGG);
  k_bn_stats<<<512, 256, 0, stream>>>(AGG, ST);
  k_enc1<<<(N_NODES * 256) / 256, 256, 0, stream>>>(AGG, X, ST, bn1_g, bn1_b, E1);
  k_gemm_enc2<<<N_NODES / 16, 256, 0, stream>>>(E1, fc1_w, fc1_b, fc2_w, fc2_b, E2);
  k_pool<<<(N_NODES * 64) / 256, 256, 0, stream>>>(E2, x_batch, PL);
  k_gemm_small<<<16, 256, 0, stream>>>(PL, mean_w, mean_b, TM);
  k_gemm_small<<<16, 256, 0, stream>>>(PL, lv_w, lv_b, out_lv);
  k_meanbn<<<1, 256, 0, stream>>>(TM, mbeta, out_mean);
  k_gemm_phi<<<N_NODES / 16, 256, 0, stream>>>(E1, PL, x_batch, phi_w, phi_b, out_phi);
  k_softmax<<<(N_NODES + 7) / 8, 256, 0, stream>>>(out_phi);
}